// InvWarpFlow_19464791786133
// MI455X (gfx1250) — compile-verified
//
#include <hip/hip_runtime.h>

// dense_image_warp(img, -flow) == bilinear sample of img at (grid + flow).
// img:  [B,H,W,C] f32, NHWC, C=32  (B=16,H=256,W=512 for this problem)
// flow: [B,H,W,2] f32
// out:  [B,H,W,C] f32
//
// Mapping: 8 lanes per pixel; each lane owns one float4 (4 channels).
// tid = pixel*8 + c4  ->  lanes 0..7 of a pixel cover the 128B channel row.
// All four bilinear neighbors are unguarded global_load_b128 (clamped floors
// guarantee in-bounds). Output is streamed with a non-temporal b128 store so
// the 256MB image stays resident in the 192MB L2 (CDNA5 TH hint).

typedef float v4f __attribute__((ext_vector_type(4)));
typedef float v2f __attribute__((ext_vector_type(2)));

#define IW_H 256
#define IW_W 512
#define IW_C 32

__global__ __launch_bounds__(256) void
InvWarpFlow_warp_kernel(const float* __restrict__ img,
                        const float* __restrict__ flow,
                        float* __restrict__ out,
                        int n_threads) {
    int tid = blockIdx.x * blockDim.x + threadIdx.x;
    if (tid >= n_threads) return;

    const int c4 = tid & 7;        // which float4 of the 32 channels
    const int p  = tid >> 3;       // linear pixel index over (b,i,j)
    const int j  = p & (IW_W - 1);
    const int bi = p >> 9;         // b*H + i   (W = 2^9)
    const int i  = bi & (IW_H - 1);
    const int b  = bi >> 8;        // H = 2^8

    // Per-pixel flow (8 lanes of a pixel hit the same 8 bytes -> L0 broadcast).
    const v2f f = ((const v2f* __restrict__)flow)[p];

    // Query point = grid + flow (reference negates flow twice).
    const float qy = (float)i + f.x;
    const float qx = (float)j + f.y;

    float fy = floorf(qy);
    float fx = floorf(qx);
    fy = fminf(fmaxf(fy, 0.0f), (float)(IW_H - 2));
    fx = fminf(fmaxf(fx, 0.0f), (float)(IW_W - 2));
    const float ay = fminf(fmaxf(qy - fy, 0.0f), 1.0f);
    const float ax = fminf(fmaxf(qx - fx, 0.0f), 1.0f);
    const int fyi = (int)fy;
    const int fxi = (int)fx;

    // float4-granular index of the top-left neighbor's first channel group.
    // img pixel stride = 8 float4s (32 floats); row stride = W*8 float4s.
    const int tlPix = (b * IW_H + fyi) * IW_W + fxi;
    const int tlIdx = tlPix * 8 + c4;

    const v4f* __restrict__ src = (const v4f* __restrict__)img;
    const v4f tl = src[tlIdx];                  // (fy  , fx  )
    const v4f tr = src[tlIdx + 8];              // (fy  , fx+1): +C floats
    const v4f bl = src[tlIdx + IW_W * 8];       // (fy+1, fx  )
    const v4f br = src[tlIdx + IW_W * 8 + 8];   // (fy+1, fx+1)

    // top = tl + (tr-tl)*ax ; bot = bl + (br-bl)*ax ; out = top + (bot-top)*ay
    v4f top, bot, o;
    top.x = __builtin_fmaf(ax, tr.x - tl.x, tl.x);
    top.y = __builtin_fmaf(ax, tr.y - tl.y, tl.y);
    top.z = __builtin_fmaf(ax, tr.z - tl.z, tl.z);
    top.w = __builtin_fmaf(ax, tr.w - tl.w, tl.w);
    bot.x = __builtin_fmaf(ax, br.x - bl.x, bl.x);
    bot.y = __builtin_fmaf(ax, br.y - bl.y, bl.y);
    bot.z = __builtin_fmaf(ax, br.z - bl.z, bl.z);
    bot.w = __builtin_fmaf(ax, br.w - bl.w, bl.w);
    o.x = __builtin_fmaf(ay, bot.x - top.x, top.x);
    o.y = __builtin_fmaf(ay, bot.y - top.y, top.y);
    o.z = __builtin_fmaf(ay, bot.z - top.z, top.z);
    o.w = __builtin_fmaf(ay, bot.w - top.w, top.w);

    // Streaming write-once output: non-temporal so img keeps the L2.
    __builtin_nontemporal_store(o, ((v4f*)out) + tid);
}

extern "C" void kernel_launch(void* const* d_in, const int* in_sizes, int n_in,
                              void* d_out, int out_size, void* d_ws, size_t ws_size,
                              hipStream_t stream) {
    const float* img  = (const float*)d_in[0];   // 16*256*512*32 f32
    const float* flow = (const float*)d_in[1];   // 16*256*512*2  f32
    float* out = (float*)d_out;

    const int n_img = in_sizes[0];               // total img elements
    const int n_threads = n_img / 4;             // one float4 per thread
    const int block = 256;
    const int grid = (n_threads + block - 1) / block;

    InvWarpFlow_warp_kernel<<<grid, block, 0, stream>>>(img, flow, out, n_threads);
}